// RoiAlign_56779467653352
// MI455X (gfx1250) — compile-verified
//
#include <hip/hip_runtime.h>

typedef __attribute__((ext_vector_type(2))) float v2f;
typedef __attribute__((ext_vector_type(8))) float v8f;

#define NBOX 1000
#define TOPK 500
#define NCLS 80
#define CROP 14
#define CH   256
#define CHUNK 32          // channels per LDS pass
#define NROW (2*CROP)     // 28 x-interpolated source rows per ROI

__device__ __forceinline__ int iclamp(int v, int lo, int hi) {
    return v < lo ? lo : (v > hi ? hi : v);
}

// ---------------- per-box score = max over classes ----------------
__global__ void k_scores(const float* __restrict__ cls, float* __restrict__ scores) {
    int i = blockIdx.x * blockDim.x + threadIdx.x;
    if (i >= NBOX) return;
    const float* p = cls + i * NCLS;
    float m = p[0];
    for (int c = 1; c < NCLS; ++c) m = fmaxf(m, p[c]);
    scores[i] = m;
}

// ---------------- exact top-k by rank counting (desc, index tiebreak) ----------------
__global__ void k_topk(const float* __restrict__ scores, int* __restrict__ topk) {
    __shared__ float s[NBOX];
    for (int j = threadIdx.x; j < NBOX; j += blockDim.x) s[j] = scores[j];
    __syncthreads();
    int i = blockIdx.x * blockDim.x + threadIdx.x;
    if (i >= NBOX) return;
    float si = s[i];
    int rank = 0;
    for (int j = 0; j < NBOX; ++j) {
        float sj = s[j];
        rank += (sj > si) || (sj == si && j < i);
    }
    if (rank < TOPK) topk[rank] = i;
}

// ---------------- FPN level per top-k box ----------------
__global__ void k_levels(const float* __restrict__ boxes, const int* __restrict__ topk,
                         int* __restrict__ lvl) {
    int p = blockIdx.x * blockDim.x + threadIdx.x;
    if (p >= TOPK) return;
    int i = topk[p];
    float x1 = boxes[i*4+0], y1 = boxes[i*4+1], x2 = boxes[i*4+2], y2 = boxes[i*4+3];
    float w = x2 - x1, h = y2 - y1;
    float size = sqrtf(w * h);
    float lv = floorf(1.0f + log2f(size / 224.0f + 1e-7f));
    lv = fminf(fmaxf(lv, 0.0f), 4.0f);
    lvl[p] = (int)lv;
}

// ---------------- stable ascending sort by level (rank counting) ----------------
__global__ void k_sort(const int* __restrict__ lvl, int* __restrict__ spos) {
    __shared__ int s[TOPK];
    for (int j = threadIdx.x; j < TOPK; j += blockDim.x) s[j] = lvl[j];
    __syncthreads();
    int p = blockIdx.x * blockDim.x + threadIdx.x;
    if (p >= TOPK) return;
    int lp = s[p];
    int rank = 0;
    for (int q = 0; q < TOPK; ++q) {
        int lq = s[q];
        rank += (lq < lp) || (lq == lp && q < p);
    }
    spos[rank] = p;
}

// ---------------- gather sorted metadata + emit sorted boxes ----------------
__global__ void k_gather_meta(const int* __restrict__ topk, const int* __restrict__ lvl,
                              const int* __restrict__ spos, const float* __restrict__ boxes,
                              int* __restrict__ soidx, int* __restrict__ slvl,
                              float* __restrict__ out_boxes) {
    int r = blockIdx.x * blockDim.x + threadIdx.x;
    if (r >= TOPK) return;
    int p = spos[r];
    int oi = topk[p];
    soidx[r] = oi;
    slvl[r]  = lvl[p];
    out_boxes[r*4+0] = boxes[oi*4+0];
    out_boxes[r*4+1] = boxes[oi*4+1];
    out_boxes[r*4+2] = boxes[oi*4+2];
    out_boxes[r*4+3] = boxes[oi*4+3];
}

// ---------------- emit sorted classification ----------------
__global__ void k_gather_cls(const float* __restrict__ cls, const int* __restrict__ soidx,
                             float* __restrict__ out_cls) {
    int t = blockIdx.x * blockDim.x + threadIdx.x;
    if (t >= TOPK * NCLS) return;
    int r = t / NCLS, c = t - r * NCLS;
    out_cls[t] = cls[soidx[r] * NCLS + c];
}

// ---------------- ROI align: VALU x-interp into LDS, WMMA y-blend ----------------
// out[y,(x,c)] = sum_r Wy[y][r] * rowX[r][(x,c)],  Wy 16x28 banded (2 nonzeros/row),
// done as 7 K=4 steps of V_WMMA_F32_16X16X4_F32 over N-tiles of 16 columns.
__global__ void __launch_bounds__(128) k_roi(
    const int* __restrict__ image_shape, const float* __restrict__ boxes,
    const float* __restrict__ f0, const float* __restrict__ f1,
    const float* __restrict__ f2, const float* __restrict__ f3,
    const float* __restrict__ f4,
    const int* __restrict__ soidx, const int* __restrict__ slvl,
    float* __restrict__ out_rois)
{
    __shared__ float rowX[NROW][CROP][CHUNK];   // 28*14*32*4 = 50176 B
    __shared__ int   sY0[CROP], sY1[CROP], sX0[CROP], sX1[CROP];
    __shared__ float sWy[CROP], sYv[CROP], sWx[CROP], sXv[CROP];

    const int roi = blockIdx.x;
    const int tid = threadIdx.x;
    const int L   = slvl[roi];
    const int oi  = soidx[roi];
    const int H   = 128 >> L;                   // square feature maps
    const float* __restrict__ F =
        (L == 0) ? f0 : (L == 1) ? f1 : (L == 2) ? f2 : (L == 3) ? f3 : f4;
    const float imgH = (float)image_shape[1];
    const float imgW = (float)image_shape[2];

    if (tid < CROP) {
        const int j = tid;
        const float fh = (float)H;
        const float x1 = boxes[oi*4+0], y1 = boxes[oi*4+1];
        const float x2 = boxes[oi*4+2], y2 = boxes[oi*4+3];
        const float t = (float)j * (1.0f / 13.0f);
        // y sampling coordinate (TF crop_and_resize chain, fh==fw)
        float a  = y1 / imgH * fh / (fh - 1.0f);
        float b  = (y2 / imgH * fh - 1.0f) / (fh - 1.0f);
        float ys = (a + (b - a) * t) * (fh - 1.0f);
        sYv[j] = (ys >= 0.0f && ys <= fh - 1.0f) ? 1.0f : 0.0f;
        float y0f = floorf(ys);
        sWy[j] = ys - y0f;
        int y0i = iclamp((int)y0f, 0, H - 1);
        sY0[j] = y0i;
        sY1[j] = iclamp(y0i + 1, 0, H - 1);
        // x sampling coordinate
        float ax = x1 / imgW * fh / (fh - 1.0f);
        float bx = (x2 / imgW * fh - 1.0f) / (fh - 1.0f);
        float xs = (ax + (bx - ax) * t) * (fh - 1.0f);
        sXv[j] = (xs >= 0.0f && xs <= fh - 1.0f) ? 1.0f : 0.0f;
        float x0f = floorf(xs);
        sWx[j] = xs - x0f;
        int x0i = iclamp((int)x0f, 0, H - 1);
        sX0[j] = x0i;
        sX1[j] = iclamp(x0i + 1, 0, H - 1);
    }
    __syncthreads();

    const int lane = tid & 31;
    const int wave = tid >> 5;          // 4 waves
    const bool lo  = lane < 16;
    const int m    = lane & 15;

    // Hoisted A-matrix (Wy): per ISA layout lane m holds K = 4s+{0,1} (lanes<16)
    // or 4s+{2,3} (lanes>=16). Each lane is nonzero at exactly one k-step
    // sAct = m>>1 (when the K-pair parity matches the lane half). Invariant
    // across tiles and channel chunks, so compute it once.
    v2f aval; aval.x = 0.0f; aval.y = 0.0f;
    int sAct = -1;
    if (m < CROP && (((m & 1) == 0) == lo)) {
        float wy = sWy[m], yv = sYv[m];
        aval.x = (1.0f - wy) * yv;      // Wy[m][2m]   (top weight)
        aval.y = wy * yv;               // Wy[m][2m+1] (bottom weight)
        sAct = m >> 1;
    }

    // Stage-1 lane remap: 8 channel-quads (float4) x 4 (r,x) pairs per wave.
    const int c4 = (lane & 7) * 4;
    const int pe = lane >> 3;           // 0..3

    float* __restrict__ outp = out_rois + (size_t)roi * (CROP * CROP * CH);

    for (int cb = 0; cb < CH; cb += CHUNK) {
        // Stage 1: x-interpolate the 28 source rows for this channel chunk.
        // 128-bit global loads + 128-bit LDS stores; per-wave trip counts uniform.
        for (int e = wave * 4 + pe; e < NROW * CROP; e += 16) {
            int r = e / CROP;
            int x = e - r * CROP;
            int rsrc = (r & 1) ? sY1[r >> 1] : sY0[r >> 1];
            const float* rb = F + ((size_t)rsrc * (size_t)H) * CH + cb + c4;
            float4 v0 = *(const float4*)(rb + (size_t)sX0[x] * CH);
            float4 v1 = *(const float4*)(rb + (size_t)sX1[x] * CH);
            float wx = sWx[x], xv = sXv[x];
            float4 o;
            o.x = xv * ((1.0f - wx) * v0.x + wx * v1.x);
            o.y = xv * ((1.0f - wx) * v0.y + wx * v1.y);
            o.z = xv * ((1.0f - wx) * v0.z + wx * v1.z);
            o.w = xv * ((1.0f - wx) * v0.w + wx * v1.w);
            *(float4*)&rowX[r][x][c4] = o;
        }
        __syncthreads();

        // Stage 2: y-blend as WMMA. 448 columns -> 28 N-tiles, 7 per wave.
        for (int tile = wave; tile < (CROP * CHUNK) / 16; tile += 4) {
            v8f acc = {};
            const int col = tile * 16 + m;       // N = lane%16 for B and D
            const int x = col >> 5;              // col / CHUNK
            const int c = col & (CHUNK - 1);
            #pragma unroll
            for (int s = 0; s < 7; ++s) {        // K = 28 in steps of 4
                v2f a;
                a.x = (s == sAct) ? aval.x : 0.0f;
                a.y = (s == sAct) ? aval.y : 0.0f;
                const int rbase = 4 * s + (lo ? 0 : 2);
                v2f b;
                b.x = rowX[rbase][x][c];
                b.y = rowX[rbase + 1][x][c];
                acc = __builtin_amdgcn_wmma_f32_16x16x4_f32(
                    false, a, false, b, (short)0, acc, false, false);
            }
            // D layout: VGPR v -> M = v + (lane<16 ? 0 : 8), N = lane%16
            #pragma unroll
            for (int v = 0; v < 8; ++v) {
                int yy = v + (lo ? 0 : 8);
                if (yy < CROP)
                    outp[((size_t)yy * CROP + x) * CH + cb + c] = acc[v];
            }
        }
        __syncthreads();  // rowX reused by next chunk's stage 1
    }
}

extern "C" void kernel_launch(void* const* d_in, const int* in_sizes, int n_in,
                              void* d_out, int out_size, void* d_ws, size_t ws_size,
                              hipStream_t stream) {
    (void)in_sizes; (void)n_in; (void)out_size; (void)ws_size;
    const int*   image_shape = (const int*)d_in[0];
    const float* boxes       = (const float*)d_in[1];
    const float* cls         = (const float*)d_in[2];
    const float* f0 = (const float*)d_in[3];
    const float* f1 = (const float*)d_in[4];
    const float* f2 = (const float*)d_in[5];
    const float* f3 = (const float*)d_in[6];
    const float* f4 = (const float*)d_in[7];

    float* out       = (float*)d_out;
    float* out_boxes = out;              // [1,500,4]
    float* out_cls   = out + 2000;       // [1,500,80]
    float* out_rois  = out + 42000;      // [1,500,14,14,256]

    char* ws = (char*)d_ws;
    float* scores = (float*)(ws + 0);        // 1000 f32
    int*   topk   = (int*)(ws + 4096);       // 500 i32
    int*   lvl    = (int*)(ws + 6144);       // 500 i32
    int*   spos   = (int*)(ws + 8192);       // 500 i32
    int*   soidx  = (int*)(ws + 10240);      // 500 i32
    int*   slvl   = (int*)(ws + 12288);      // 500 i32

    k_scores<<<(NBOX + 255) / 256, 256, 0, stream>>>(cls, scores);
    k_topk<<<(NBOX + 255) / 256, 256, 0, stream>>>(scores, topk);
    k_levels<<<(TOPK + 255) / 256, 256, 0, stream>>>(boxes, topk, lvl);
    k_sort<<<(TOPK + 255) / 256, 256, 0, stream>>>(lvl, spos);
    k_gather_meta<<<(TOPK + 255) / 256, 256, 0, stream>>>(topk, lvl, spos, boxes,
                                                          soidx, slvl, out_boxes);
    k_gather_cls<<<(TOPK * NCLS + 255) / 256, 256, 0, stream>>>(cls, soidx, out_cls);
    k_roi<<<TOPK, 128, 0, stream>>>(image_shape, boxes, f0, f1, f2, f3, f4,
                                    soidx, slvl, out_rois);
}